// GCN_22093311771207
// MI455X (gfx1250) — compile-verified
//
#include <hip/hip_runtime.h>
#include <hip/hip_bf16.h>
#include <math.h>

// GraphSAGE forward for MI455X (gfx1250), wave32, WMMA f32 16x16x4.
//
// dims
#define IN_CH 128
#define H1CH  32
#define H2CH  20
#define NCLS  10

typedef float v2f __attribute__((ext_vector_type(2)));
typedef float v8f __attribute__((ext_vector_type(8)));

// ---------------------------------------------------------------- zero fill
__global__ void zero_kernel(float* __restrict__ p, long long n) {
    long long i = (long long)blockIdx.x * blockDim.x + threadIdx.x;
    long long stride = (long long)gridDim.x * blockDim.x;
    for (; i < n; i += stride) p[i] = 0.0f;
}

// ---------------------------------------------------------------- degree
__global__ void deg_kernel(const int* __restrict__ dst, float* __restrict__ deg, int nE) {
    int e = blockIdx.x * blockDim.x + threadIdx.x;
    if (e < nE) atomicAdd(&deg[dst[e]], 1.0f);
}

// ------------------------------------------------- scatter x[src] -> agg[dst], F=128
__global__ void scatter128_kernel(const float* __restrict__ x, const int* __restrict__ src,
                                  const int* __restrict__ dst, float* __restrict__ agg, int nE) {
    long long t = (long long)blockIdx.x * blockDim.x + threadIdx.x;
    int e = (int)(t >> 5);            // 32 threads per edge, 4 floats each
    if (e >= nE) return;
    int c = ((int)t & 31) * 4;
    int s = src[e], d = dst[e];
    const float4 v = *(const float4*)(x + (size_t)s * IN_CH + c);
    float* o = agg + (size_t)d * IN_CH + c;
    atomicAdd(o + 0, v.x); atomicAdd(o + 1, v.y);
    atomicAdd(o + 2, v.z); atomicAdd(o + 3, v.w);
}

// ------------------------------------------------- scatter h[src] -> agg[dst], F=32
__global__ void scatter32_kernel(const float* __restrict__ h, const int* __restrict__ src,
                                 const int* __restrict__ dst, float* __restrict__ agg, int nE) {
    long long t = (long long)blockIdx.x * blockDim.x + threadIdx.x;
    int e = (int)(t >> 3);            // 8 threads per edge, 4 floats each
    if (e >= nE) return;
    int c = ((int)t & 7) * 4;
    int s = src[e], d = dst[e];
    const float4 v = *(const float4*)(h + (size_t)s * H1CH + c);
    float* o = agg + (size_t)d * H1CH + c;
    atomicAdd(o + 0, v.x); atomicAdd(o + 1, v.y);
    atomicAdd(o + 2, v.z); atomicAdd(o + 3, v.w);
}

// ---------------------------------------------------------------- SAGE layer 1
// h1 = relu( l2norm( (agg/deg) @ Wl + bl + x @ Wr ) ), 128 -> 32
// One wave computes a 16(node) x 32(feature) tile via two 16x16 accumulators.
__global__ void sage1_kernel(const float* __restrict__ x, const float* __restrict__ agg,
                             const float* __restrict__ deg,
                             const float* __restrict__ Wl, const float* __restrict__ bl,
                             const float* __restrict__ Wr,
                             float* __restrict__ h, int nNodes) {
    const int wid   = (blockIdx.x * blockDim.x + threadIdx.x) >> 5;
    const int lane  = threadIdx.x & 31;
    const int col   = lane & 15;      // A: row index; B/C/D: column index
    const int half  = lane >> 4;      // selects K pair (A/B) / row group (C/D)
    const int mbase = wid * 16;

    int nrow = mbase + col; if (nrow >= nNodes) nrow = nNodes - 1;
    const float dscale = 1.0f / fmaxf(deg[nrow], 1.0f);

    v8f acc0 = {}; v8f acc1 = {};
    const float* arow = agg + (size_t)nrow * IN_CH + 2 * half;
    const float* xrow = x   + (size_t)nrow * IN_CH + 2 * half;

    // (agg/deg) @ Wl
    #pragma unroll 4
    for (int k = 0; k < IN_CH; k += 4) {
        const float2 av = *(const float2*)(arow + k);
        v2f a; a.x = av.x * dscale; a.y = av.y * dscale;
        const float* wb = Wl + (size_t)(k + 2 * half) * H1CH;
        v2f b0; b0.x = wb[col];        b0.y = wb[H1CH + col];
        v2f b1; b1.x = wb[16 + col];   b1.y = wb[H1CH + 16 + col];
        acc0 = __builtin_amdgcn_wmma_f32_16x16x4_f32(false, a, false, b0, (short)0, acc0, false, false);
        acc1 = __builtin_amdgcn_wmma_f32_16x16x4_f32(false, a, false, b1, (short)0, acc1, false, false);
    }
    // x @ Wr
    #pragma unroll 4
    for (int k = 0; k < IN_CH; k += 4) {
        const float2 xv = *(const float2*)(xrow + k);
        v2f a; a.x = xv.x; a.y = xv.y;
        const float* wb = Wr + (size_t)(k + 2 * half) * H1CH;
        v2f b0; b0.x = wb[col];        b0.y = wb[H1CH + col];
        v2f b1; b1.x = wb[16 + col];   b1.y = wb[H1CH + 16 + col];
        acc0 = __builtin_amdgcn_wmma_f32_16x16x4_f32(false, a, false, b0, (short)0, acc0, false, false);
        acc1 = __builtin_amdgcn_wmma_f32_16x16x4_f32(false, a, false, b1, (short)0, acc1, false, false);
    }

    const float bias0 = bl[col];
    const float bias1 = bl[16 + col];
    #pragma unroll
    for (int i = 0; i < 8; ++i) { acc0[i] += bias0; acc1[i] += bias1; }

    // per-row L2 norm: row i+8*half lives across the 16 lanes of this half
    #pragma unroll
    for (int i = 0; i < 8; ++i) {
        float s = acc0[i] * acc0[i] + acc1[i] * acc1[i];
        s += __shfl_xor(s, 1);
        s += __shfl_xor(s, 2);
        s += __shfl_xor(s, 4);
        s += __shfl_xor(s, 8);
        const float sc = 1.0f / fmaxf(sqrtf(s), 1e-12f);
        acc0[i] = fmaxf(acc0[i] * sc, 0.0f);   // ReLU fused
        acc1[i] = fmaxf(acc1[i] * sc, 0.0f);
    }

    #pragma unroll
    for (int i = 0; i < 8; ++i) {
        const int m = mbase + i + 8 * half;
        if (m < nNodes) {
            h[(size_t)m * H1CH + col]      = acc0[i];
            h[(size_t)m * H1CH + 16 + col] = acc1[i];
        }
    }
}

// ---------------------------------------------------------------- SAGE layer 2
// h2 = l2norm( (agg/deg) @ Wl + bl + h1 @ Wr ), 32 -> 20 (N padded to 32 with zeros)
__global__ void sage2_kernel(const float* __restrict__ hin, const float* __restrict__ agg,
                             const float* __restrict__ deg,
                             const float* __restrict__ Wl, const float* __restrict__ bl,
                             const float* __restrict__ Wr,
                             float* __restrict__ h2, int nNodes) {
    const int wid   = (blockIdx.x * blockDim.x + threadIdx.x) >> 5;
    const int lane  = threadIdx.x & 31;
    const int col   = lane & 15;
    const int half  = lane >> 4;
    const int mbase = wid * 16;

    int nrow = mbase + col; if (nrow >= nNodes) nrow = nNodes - 1;
    const float dscale = 1.0f / fmaxf(deg[nrow], 1.0f);

    v8f acc0 = {}; v8f acc1 = {};
    const float* arow = agg + (size_t)nrow * H1CH + 2 * half;
    const float* xrow = hin + (size_t)nrow * H1CH + 2 * half;
    const bool hi_ok = (col < (H2CH - 16));   // columns 16..19 valid in second tile

    #pragma unroll
    for (int k = 0; k < H1CH; k += 4) {
        const float2 av = *(const float2*)(arow + k);
        v2f a; a.x = av.x * dscale; a.y = av.y * dscale;
        const float* wb = Wl + (size_t)(k + 2 * half) * H2CH;
        v2f b0; b0.x = wb[col];                       b0.y = wb[H2CH + col];
        v2f b1; b1.x = hi_ok ? wb[16 + col] : 0.0f;   b1.y = hi_ok ? wb[H2CH + 16 + col] : 0.0f;
        acc0 = __builtin_amdgcn_wmma_f32_16x16x4_f32(false, a, false, b0, (short)0, acc0, false, false);
        acc1 = __builtin_amdgcn_wmma_f32_16x16x4_f32(false, a, false, b1, (short)0, acc1, false, false);
    }
    #pragma unroll
    for (int k = 0; k < H1CH; k += 4) {
        const float2 xv = *(const float2*)(xrow + k);
        v2f a; a.x = xv.x; a.y = xv.y;
        const float* wb = Wr + (size_t)(k + 2 * half) * H2CH;
        v2f b0; b0.x = wb[col];                       b0.y = wb[H2CH + col];
        v2f b1; b1.x = hi_ok ? wb[16 + col] : 0.0f;   b1.y = hi_ok ? wb[H2CH + 16 + col] : 0.0f;
        acc0 = __builtin_amdgcn_wmma_f32_16x16x4_f32(false, a, false, b0, (short)0, acc0, false, false);
        acc1 = __builtin_amdgcn_wmma_f32_16x16x4_f32(false, a, false, b1, (short)0, acc1, false, false);
    }

    const float bias0 = bl[col];
    const float bias1 = hi_ok ? bl[16 + col] : 0.0f;
    #pragma unroll
    for (int i = 0; i < 8; ++i) { acc0[i] += bias0; acc1[i] += bias1; }

    #pragma unroll
    for (int i = 0; i < 8; ++i) {
        float s = acc0[i] * acc0[i] + acc1[i] * acc1[i];  // padded cols are exactly 0
        s += __shfl_xor(s, 1);
        s += __shfl_xor(s, 2);
        s += __shfl_xor(s, 4);
        s += __shfl_xor(s, 8);
        const float sc = 1.0f / fmaxf(sqrtf(s), 1e-12f);
        acc0[i] *= sc;                                     // no ReLU on layer 2
        acc1[i] *= sc;
    }

    #pragma unroll
    for (int i = 0; i < 8; ++i) {
        const int m = mbase + i + 8 * half;
        if (m < nNodes) {
            h2[(size_t)m * H2CH + col] = acc0[i];
            if (hi_ok) h2[(size_t)m * H2CH + 16 + col] = acc1[i];
        }
    }
}

// ---------------------------------------------------------------- logits + softmax
__global__ void out_kernel(const float* __restrict__ h2, const float* __restrict__ W,
                           const float* __restrict__ b, float* __restrict__ out, int nNodes) {
    int n = blockIdx.x * blockDim.x + threadIdx.x;
    if (n >= nNodes) return;
    float f[H2CH];
    #pragma unroll
    for (int k = 0; k < H2CH; ++k) f[k] = h2[(size_t)n * H2CH + k];
    float lg[NCLS];
    #pragma unroll
    for (int j = 0; j < NCLS; ++j) {
        float s = b[j];
        #pragma unroll
        for (int k = 0; k < H2CH; ++k) s += f[k] * W[k * NCLS + j];
        lg[j] = s;
    }
    float mx = lg[0];
    #pragma unroll
    for (int j = 1; j < NCLS; ++j) mx = fmaxf(mx, lg[j]);
    float sum = 0.0f;
    #pragma unroll
    for (int j = 0; j < NCLS; ++j) { lg[j] = __expf(lg[j] - mx); sum += lg[j]; }
    const float inv = 1.0f / sum;
    #pragma unroll
    for (int j = 0; j < NCLS; ++j) out[(size_t)n * NCLS + j] = lg[j] * inv;
}

// ---------------------------------------------------------------- launch
extern "C" void kernel_launch(void* const* d_in, const int* in_sizes, int n_in,
                              void* d_out, int out_size, void* d_ws, size_t ws_size,
                              hipStream_t stream) {
    (void)n_in; (void)out_size; (void)ws_size;
    const float* x   = (const float*)d_in[0];
    const int*   ei  = (const int*)d_in[1];
    const float* W1l = (const float*)d_in[2];
    const float* b1l = (const float*)d_in[3];
    const float* W1r = (const float*)d_in[4];
    const float* W2l = (const float*)d_in[5];
    const float* b2l = (const float*)d_in[6];
    const float* W2r = (const float*)d_in[7];
    const float* Wo  = (const float*)d_in[8];
    const float* bo  = (const float*)d_in[9];

    const int nN = in_sizes[0] / IN_CH;
    const int nE = in_sizes[1] / 2;
    const int* src = ei;
    const int* dst = ei + nE;

    // workspace layout (floats): deg | agg1 | agg2 | h1 | h2
    float* ws   = (float*)d_ws;
    float* deg  = ws;
    float* agg1 = deg  + nN;
    float* agg2 = agg1 + (size_t)nN * IN_CH;
    float* h1   = agg2 + (size_t)nN * H1CH;
    float* h2   = h1   + (size_t)nN * H1CH;

    // zero deg + agg1 + agg2 (contiguous)
    const long long zn = (long long)nN * (1 + IN_CH + H1CH);
    zero_kernel<<<2048, 256, 0, stream>>>(ws, zn);

    deg_kernel<<<(nE + 255) / 256, 256, 0, stream>>>(dst, deg, nE);

    const long long t1 = (long long)nE * (IN_CH / 4);
    scatter128_kernel<<<(int)((t1 + 255) / 256), 256, 0, stream>>>(x, src, dst, agg1, nE);

    const int waves  = (nN + 15) / 16;
    const int blocks = (waves + 1) / 2;     // 2 waves (64 threads) per block
    sage1_kernel<<<blocks, 64, 0, stream>>>(x, agg1, deg, W1l, b1l, W1r, h1, nN);

    const long long t2 = (long long)nE * (H1CH / 4);
    scatter32_kernel<<<(int)((t2 + 255) / 256), 256, 0, stream>>>(h1, src, dst, agg2, nE);

    sage2_kernel<<<blocks, 64, 0, stream>>>(h1, agg2, deg, W2l, b2l, W2r, h2, nN);

    out_kernel<<<(nN + 255) / 256, 256, 0, stream>>>(h2, Wo, bo, (float*)d_out, nN);
}